// SecondLayer_70042326663885
// MI455X (gfx1250) — compile-verified
//
#include <hip/hip_runtime.h>
#include <hip/hip_bf16.h>

// Problem constants (match reference).
#define N_D_FEAT 256   // K dimension of dense GEMM
#define N_UNITS  128   // output features

typedef __attribute__((ext_vector_type(2))) float v2f;
typedef __attribute__((ext_vector_type(8))) float v8f;
typedef __attribute__((address_space(3))) float lds_float;

// ---------------------------------------------------------------------------
// Kernel 1: h = x @ w  via V_WMMA_F32_16X16X4_F32 (exact fp32, no downcast).
//
// Grid:  n_nodes/16 blocks, 256 threads (8 waves) per block.
// Block: covers 16 rows of x; wave i computes the 16x16 output tile at
//        columns [16*i, 16*i+16).  The 16x256 fp32 strip of x (16 KB) is
//        staged into LDS with gfx1250 async-to-LDS loads (ASYNCcnt path),
//        tagged TH_LOAD_NT so the once-read x stream (102 MB) does not evict
//        the L2-resident h/out working set needed by the scatter phase.
//
// Fragment layouts per CDNA5 ISA 7.12.2:
//   A (16x4 f32, 2 VGPRs): lanes 0-15 -> M=lane,    K = {0,1} in v[0],v[1]
//                          lanes 16-31 -> M=lane-16, K = {2,3} in v[0],v[1]
//   B (4x16 f32, 2 VGPRs): lanes 0-15 -> N=lane,    K = {0,1} in v[0],v[1]
//                          lanes 16-31 -> N=lane-16, K = {2,3}
//   C/D (16x16 f32, 8 VGPRs): VGPR v: lanes 0-15 -> M=v, lanes 16-31 -> M=v+8,
//                             N = lane & 15.
// ---------------------------------------------------------------------------
__global__ __launch_bounds__(256) void gat_gemm_wmma(
    const float* __restrict__ x,   // [n_nodes, 256]
    const float* __restrict__ w,   // [256, 128]
    float* __restrict__ h,         // [n_nodes, 128]  (workspace)
    int n_nodes) {
  __shared__ float xs[16 * N_D_FEAT];  // 16 KB

  const int m16 = blockIdx.x * 16;     // first row of this block's strip

  // Async-stage 16 rows of x (1024 x 16B) into LDS, non-temporal in L2.
  // The LDS byte offset for the async op's vdst is obtained by an
  // addrspace(3) cast + ptrtoint (AS3 pointers are 32-bit LDS offsets).
  // Passing the LDS-derived offset and xs itself into the volatile asm with
  // a "memory" clobber marks xs as captured/written, so the compiler keeps
  // the LDS object and the later ds_load reads (an earlier revision lost
  // both because only a plain integer escaped into the asm).
  {
    lds_float* xs3 = (lds_float*)xs;   // generic -> addrspace(3)
    const float4* src = (const float4*)(x + (size_t)m16 * N_D_FEAT);
    for (int i = threadIdx.x; i < (16 * N_D_FEAT) / 4; i += blockDim.x) {
      unsigned lds_off = (unsigned)(size_t)(xs3 + i * 4);
      const float4* gp = src + i;
      asm volatile(
          "global_load_async_to_lds_b128 %0, %1, off th:TH_LOAD_NT"
          :
          : "v"(lds_off), "v"(gp), "s"(xs)
          : "memory");
    }
    asm volatile("s_wait_asynccnt 0x0" ::: "memory");
  }
  __syncthreads();

  const int wave = threadIdx.x >> 5;   // 0..7 -> column tile
  const int lane = threadIdx.x & 31;
  const int half = lane >> 4;          // 0: K lo pair, 1: K hi pair
  const int l16  = lane & 15;          // M (for A) / N (for B,D)
  const int n16  = wave * 16;

  // LDS row base for this lane's A rows; consecutive K pair -> ds_load_b64.
  const float2* arow = (const float2*)(xs + l16 * N_D_FEAT);

  v8f c = {};
  #pragma unroll 4
  for (int k = 0; k < N_D_FEAT; k += 4) {
    // A fragment: x[m16+l16][k + 2*half + {0,1}]
    float2 af = arow[(k >> 1) + half];
    v2f a;  a[0] = af.x;  a[1] = af.y;

    // B fragment: w[k + 2*half + {0,1}][n16 + l16] (128 KB matrix, WGP$/L2
    // resident, high reuse -> keep regular-temporal)
    const float* wp = w + (size_t)(k + 2 * half) * N_UNITS + n16 + l16;
    v2f b;  b[0] = wp[0];  b[1] = wp[N_UNITS];

    // D = A*B + C   (8 args: neg_a, A, neg_b, B, c_mod, C, reuse_a, reuse_b)
    c = __builtin_amdgcn_wmma_f32_16x16x4_f32(
        false, a, false, b, (short)0, c, false, false);
  }

  // Write back D tile: VGPR v -> row m16 + v + 8*half, col n16 + l16.
  // Regular-temporal: h must stay in L2 for the scatter phase.
  float* hout = h + (size_t)(m16 + 8 * half) * N_UNITS + n16 + l16;
  #pragma unroll
  for (int v = 0; v < 8; ++v)
    hout[(size_t)v * N_UNITS] = c[v];
}

// ---------------------------------------------------------------------------
// Kernel 2: zero-init out (harness poisons d_out; we must produce exact sums).
// Regular-temporal stores: primes the out lines into L2 for the atomics.
// ---------------------------------------------------------------------------
__global__ void gat_zero(float4* __restrict__ out, long long n4) {
  long long i = (long long)blockIdx.x * blockDim.x + threadIdx.x;
  if (i < n4) out[i] = make_float4(0.f, 0.f, 0.f, 0.f);
}

// ---------------------------------------------------------------------------
// Kernel 3: edge scatter  out[dst] += val * h[src]
//
// One wave (32 lanes) per edge, 4 consecutive floats per lane:
//   - gather h[src] as one coalesced global_load_b128 per lane (512B/edge)
//   - 4 non-returning global_atomic_add_f32 per lane (512B/edge)
// h (51.2 MB) and out (51.2 MB) both fit in the 192 MB L2 and the f32
// atomics execute at L2, so this phase runs near L2 bandwidth.  The edge
// arrays are read exactly once -> non-temporal loads.
// ---------------------------------------------------------------------------
__global__ __launch_bounds__(256) void gat_scatter(
    const float* __restrict__ h,
    const int*   __restrict__ adj_src,
    const int*   __restrict__ adj_dst,
    const float* __restrict__ adj_vals,
    float* __restrict__ out,
    long long n_edges) {
  long long gid = (long long)blockIdx.x * blockDim.x + threadIdx.x;
  long long e = gid >> 5;        // edge index (one wave per edge)
  int lane = threadIdx.x & 31;   // lane -> features [4*lane, 4*lane+4)
  if (e >= n_edges) return;

  int   src = __builtin_nontemporal_load(adj_src + e);
  int   dst = __builtin_nontemporal_load(adj_dst + e);
  float val = __builtin_nontemporal_load(adj_vals + e);

  const float4* hp = (const float4*)(h + (size_t)src * N_UNITS) + lane;
  float4 hv = *hp;               // coalesced 512B gather per edge

  float* op = out + (size_t)dst * N_UNITS + lane * 4;
  unsafeAtomicAdd(op + 0, val * hv.x);
  unsafeAtomicAdd(op + 1, val * hv.y);
  unsafeAtomicAdd(op + 2, val * hv.z);
  unsafeAtomicAdd(op + 3, val * hv.w);
}

// ---------------------------------------------------------------------------
extern "C" void kernel_launch(void* const* d_in, const int* in_sizes, int n_in,
                              void* d_out, int out_size, void* d_ws, size_t ws_size,
                              hipStream_t stream) {
  const float* x        = (const float*)d_in[0];  // [N, 256] f32
  const float* w        = (const float*)d_in[1];  // [256, 128] f32
  const int*   adj_src  = (const int*)d_in[2];    // [E] int32
  const int*   adj_dst  = (const int*)d_in[3];    // [E] int32
  const float* adj_vals = (const float*)d_in[4];  // [E] f32
  float* out = (float*)d_out;                     // [N, 128] f32

  const int n_nodes = in_sizes[0] / N_D_FEAT;     // 100000
  const long long n_edges = in_sizes[2];          // 1600000

  float* h = (float*)d_ws;                        // [N, 128] f32 (51.2 MB)

  // Phase 1: dense projection with fp32 WMMA.  N is a multiple of 16.
  gat_gemm_wmma<<<n_nodes / 16, 256, 0, stream>>>(x, w, h, n_nodes);

  // Phase 2: zero the accumulator (also primes out into L2).
  long long n4 = (long long)n_nodes * N_UNITS / 4;
  gat_zero<<<(unsigned)((n4 + 255) / 256), 256, 0, stream>>>((float4*)out, n4);

  // Phase 3: one wave per edge, coalesced gathers + L2-resident f32 atomics.
  long long total = n_edges * 32;
  gat_scatter<<<(unsigned)((total + 255) / 256), 256, 0, stream>>>(
      h, adj_src, adj_dst, adj_vals, out, n_edges);
}